// GatedRelevanceNetwork_335007449190
// MI455X (gfx1250) — compile-verified
//
#include <hip/hip_runtime.h>
#include <hip/hip_bf16.h>

typedef __attribute__((ext_vector_type(2))) float v2f;
typedef __attribute__((ext_vector_type(8))) float v8f;

#define EMB 256
#define KDIM 32
#define LSEQ 64

// ---------------------------------------------------------------------------
// Kernel 1: one workgroup per (b,k).
//   GEMM1: P[64x256] = arg1[b] (64x256) @ Wb[k] (256x256)   -> LDS (swizzled)
//   GEMM2: QT[64x64] = arg2[b] (64x256) @ P^T                -> btpT[bk][n][m]
// Uses V_WMMA_F32_16X16X4_F32 throughout (fp32 precision preserved).
// ---------------------------------------------------------------------------
__global__ void grn_bilinear_kernel(const float* __restrict__ arg1,
                                    const float* __restrict__ arg2,
                                    const float* __restrict__ Wb,
                                    float* __restrict__ btpT) {
    extern __shared__ float lds_P[];   // 64 rows x 256 cols fp32, XOR-swizzled, 64KB

    const int bk   = blockIdx.x;       // 0..255
    const int bb   = bk >> 5;          // batch index
    const int kk   = bk & 31;          // bilinear slice index
    const int tid  = threadIdx.x;
    const int wave = tid >> 5;         // 0..7 (wave32)
    const int lane = tid & 31;
    const int l15  = lane & 15;
    const int half = lane >> 4;        // selects K-pair {0,1} vs {2,3}

    // ------------------- GEMM1: P = arg1[b] @ Wb[k] -------------------
    const float* A1 = arg1 + (size_t)bb * LSEQ * EMB;
    const float* B1 = Wb   + (size_t)kk * EMB * EMB;

    for (int t8 = 0; t8 < 8; ++t8) {
        const int tile = wave * 8 + t8;          // 0..63 : 4 m-tiles x 16 f-tiles
        const int m0 = (tile >> 4) << 4;
        const int f0 = (tile & 15) << 4;
        const float* arow = A1 + (size_t)(m0 + l15) * EMB;  // A: lane holds row m
        const float* bcol = B1 + (f0 + l15);                // B: lane holds col f

        v8f acc = {0.f, 0.f, 0.f, 0.f, 0.f, 0.f, 0.f, 0.f};
        for (int e = 0; e < EMB; e += 4) {
            const int e2 = e + half * 2;
            v2f a = *(const v2f*)(arow + e2);               // aligned float2
            v2f bv;
            bv.x = bcol[(size_t)e2 * EMB];
            bv.y = bcol[(size_t)(e2 + 1) * EMB];
            acc = __builtin_amdgcn_wmma_f32_16x16x4_f32(
                      false, a, false, bv, (short)0, acc, false, false);
        }
        // D layout: lane -> col f = f0+l15 ; vgpr r -> row m = m0 + half*8 + r
        const int fcol = f0 + l15;
#pragma unroll
        for (int r = 0; r < 8; ++r) {
            const int mrow = m0 + half * 8 + r;
            lds_P[(mrow << 8) + (fcol ^ ((mrow & 15) << 2))] = acc[r];
        }
    }

    __syncthreads();

    // --------------- GEMM2: QT[n,m] = sum_f arg2[b][n][f] * P[m][f] ---------------
    const float* A2     = arg2 + (size_t)bb * LSEQ * EMB;
    float*       out_bk = btpT + (size_t)bk * LSEQ * LSEQ;

    for (int t2 = 0; t2 < 2; ++t2) {
        const int tile = wave * 2 + t2;          // 0..15 : 4 n-tiles x 4 m-tiles
        const int n0 = (tile >> 2) << 4;
        const int m0 = (tile & 3) << 4;
        const float* arow = A2 + (size_t)(n0 + l15) * EMB;  // A: lane holds row n
        const int mcol = m0 + l15;                          // B: lane holds col m

        v8f acc = {0.f, 0.f, 0.f, 0.f, 0.f, 0.f, 0.f, 0.f};
        for (int f = 0; f < EMB; f += 4) {
            const int f2 = f + half * 2;
            v2f a = *(const v2f*)(arow + f2);
            v2f bv;                                         // B[f][m] = P[m][f]
            bv.x = lds_P[(mcol << 8) + ( f2      ^ ((mcol & 15) << 2))];
            bv.y = lds_P[(mcol << 8) + ((f2 + 1) ^ ((mcol & 15) << 2))];
            acc = __builtin_amdgcn_wmma_f32_16x16x4_f32(
                      false, a, false, bv, (short)0, acc, false, false);
        }
#pragma unroll
        for (int r = 0; r < 8; ++r) {
            const int nrow = n0 + half * 8 + r;
            out_bk[(nrow << 6) + mcol] = acc[r];            // coalesced over lanes
        }
    }
}

// ---------------------------------------------------------------------------
// Kernel 2: the four d->K projections (a1d, a1g, a2d, a2g), one dot/thread.
// ---------------------------------------------------------------------------
__global__ void grn_proj_kernel(const float* __restrict__ arg1,
                                const float* __restrict__ arg2,
                                const float* __restrict__ Wd,
                                const float* __restrict__ Wg,
                                float* __restrict__ a1d, float* __restrict__ a1g,
                                float* __restrict__ a2d, float* __restrict__ a2g) {
    const int t   = blockIdx.x * blockDim.x + threadIdx.x;  // 65536 threads
    const int k   = t & 31;
    const int row = (t >> 5) & 511;                          // b*64 + seq
    const int arr = t >> 14;                                 // 0..3

    const float* x = ((arr < 2) ? arg1 : arg2) + (size_t)row * EMB;
    const float* W = ((arr == 0 || arr == 2) ? Wd : Wg) + ((arr >= 2) ? EMB * KDIM : 0);

    float acc = 0.f;
    for (int e = 0; e < EMB; ++e)
        acc += x[e] * W[e * KDIM + k];

    float* dst = (arr == 0) ? a1d : (arr == 1) ? a1g : (arr == 2) ? a2d : a2g;
    dst[row * KDIM + k] = acc;
}

// ---------------------------------------------------------------------------
// Kernel 3: gating + reduction over k with u.
// ---------------------------------------------------------------------------
__global__ void grn_fuse_kernel(const float* __restrict__ btpT,
                                const float* __restrict__ a1d, const float* __restrict__ a1g,
                                const float* __restrict__ a2d, const float* __restrict__ a2g,
                                const float* __restrict__ bg,  const float* __restrict__ bias,
                                const float* __restrict__ u,
                                float* __restrict__ out) {
    const int t  = blockIdx.x * blockDim.x + threadIdx.x;   // 32768 threads
    const int m  = t & 63;                                   // fastest -> coalesced btpT reads
    const int n  = (t >> 6) & 63;
    const int bb = t >> 12;

    const float* btp_b = btpT + ((size_t)bb * KDIM) * (LSEQ * LSEQ) + (n << 6) + m;
    const float* p1d = a1d + (((bb << 6) + m) * KDIM);
    const float* p2d = a2d + (((bb << 6) + n) * KDIM);
    const float* p1g = a1g + (((bb << 6) + m) * KDIM);
    const float* p2g = a2g + (((bb << 6) + n) * KDIM);

    float acc = 0.f;
#pragma unroll 4
    for (int k = 0; k < KDIM; ++k) {
        const float bt = btp_b[(size_t)k * (LSEQ * LSEQ)];
        const float xg = p1g[k] + p2g[k] + bg[k];
        const float g  = 1.0f / (1.0f + expf(-xg));
        const float s  = tanhf(p1d[k] + p2d[k]);
        const float fused = g * bt + (1.0f - g) * s + bias[k];
        acc += fused * u[k];
    }
    out[(((bb << 6) + m) << 6) + n] = acc;
}

// ---------------------------------------------------------------------------
extern "C" void kernel_launch(void* const* d_in, const int* in_sizes, int n_in,
                              void* d_out, int out_size, void* d_ws, size_t ws_size,
                              hipStream_t stream) {
    const float* arg1 = (const float*)d_in[0];
    const float* arg2 = (const float*)d_in[1];
    const float* Wb   = (const float*)d_in[2];
    const float* Wd   = (const float*)d_in[3];
    const float* Wg   = (const float*)d_in[4];
    const float* bg   = (const float*)d_in[5];
    const float* bias = (const float*)d_in[6];
    const float* u    = (const float*)d_in[7];
    float* out = (float*)d_out;

    // Workspace layout (floats): btpT[8*32][64][64], then 4 x (512*32) projections.
    float* ws   = (float*)d_ws;
    float* btpT = ws;
    float* a1d  = ws + (size_t)8 * KDIM * LSEQ * LSEQ;   // +1,048,576
    float* a1g  = a1d + 512 * KDIM;
    float* a2d  = a1g + 512 * KDIM;
    float* a2g  = a2d + 512 * KDIM;

    grn_bilinear_kernel<<<8 * KDIM, 256, LSEQ * EMB * sizeof(float), stream>>>(
        arg1, arg2, Wb, btpT);
    grn_proj_kernel<<<256, 256, 0, stream>>>(arg1, arg2, Wd, Wg, a1d, a1g, a2d, a2g);
    grn_fuse_kernel<<<128, 256, 0, stream>>>(btpT, a1d, a1g, a2d, a2g, bg, bias, u, out);
}